// CrossCCAttention_45303315038453
// MI455X (gfx1250) — compile-verified
//
#include <hip/hip_runtime.h>
#include <math.h>

// Problem constants (match reference)
#define BB     8
#define NQ     300
#define CC     256
#define HH     64
#define WW     64
#define NHEADS 8
#define HD     32
#define HWSZ   (HH * WW)   // 4096

typedef __attribute__((ext_vector_type(2))) float v2f;
typedef __attribute__((ext_vector_type(8))) float v8f;

// D = A(16x4,f32) * B(4x16,f32) + C(16x16,f32), wave32
__device__ __forceinline__ v8f wmma4(v2f a, v2f b, v8f c) {
  return __builtin_amdgcn_wmma_f32_16x16x4_f32(false, a, false, b, (short)0, c,
                                               false, false);
}

// ---------------------------------------------------------------------------
// Kernel A: pk[b,o,hw] = sum_c Wk[o,c]*key[b,c,hw] + bk[o]   (and same for V)
// One wave per block; wave computes a 16(M=o) x 64(N=hw) tile.
// grid: x = HW/64 (=64), y = C/16 (=16), z = b*2 + {0:K, 1:V}
// ---------------------------------------------------------------------------
__global__ void __launch_bounds__(32) proj_khw_kernel(
    const float* __restrict__ Wk, const float* __restrict__ bk,
    const float* __restrict__ Wv, const float* __restrict__ bv,
    const float* __restrict__ key, const float* __restrict__ val,
    float* __restrict__ pk, float* __restrict__ pv) {
  const int zz  = blockIdx.z;
  const int b   = zz >> 1;
  const bool isV = (zz & 1) != 0;
  const float* Wm   = isV ? Wv : Wk;
  const float* bias = isV ? bv : bk;
  const float* X = (isV ? val : key) + (size_t)b * CC * HWSZ;
  float*       O = (isV ? pv : pk)   + (size_t)b * CC * HWSZ;

  const int n0   = blockIdx.x * 64;
  const int m0   = blockIdx.y * 16;
  const int lane = threadIdx.x & 31;
  const int half = lane >> 4;   // 0: K={0,1}, 1: K={2,3}
  const int mr   = lane & 15;

  v8f acc0 = {0.f,0.f,0.f,0.f,0.f,0.f,0.f,0.f};
  v8f acc1 = acc0, acc2 = acc0, acc3 = acc0;

  for (int k0 = 0; k0 < CC; k0 += 4) {
    // A fragment: row m0+mr, K = k0 + 2*half + {0,1}
    v2f af = *(const v2f*)(Wm + (size_t)(m0 + mr) * CC + k0 + 2 * half);
    // B fragments: rows K = k0 + 2*half + {0,1}, cols n0+mr (+16,+32,+48)
    const float* Bp = X + (size_t)(k0 + 2 * half) * HWSZ + n0 + mr;
    v2f b0, b1, b2, b3;
    b0.x = Bp[0];   b0.y = Bp[HWSZ];
    b1.x = Bp[16];  b1.y = Bp[HWSZ + 16];
    b2.x = Bp[32];  b2.y = Bp[HWSZ + 32];
    b3.x = Bp[48];  b3.y = Bp[HWSZ + 48];
    acc0 = wmma4(af, b0, acc0);
    acc1 = wmma4(af, b1, acc1);
    acc2 = wmma4(af, b2, acc2);
    acc3 = wmma4(af, b3, acc3);
  }

  // D layout: element r -> row m0 + r + 8*half, col n0 + (lane&15)
  const int n  = n0 + mr;
  const int mb = m0 + half * 8;
#pragma unroll
  for (int r = 0; r < 8; ++r) {
    const float brow = bias[mb + r];
    float* Orow = O + (size_t)(mb + r) * HWSZ + n;
    Orow[0]  = acc0[r] + brow;
    Orow[16] = acc1[r] + brow;
    Orow[32] = acc2[r] + brow;
    Orow[48] = acc3[r] + brow;
  }
}

// ---------------------------------------------------------------------------
// Kernel B/D: Out[b,m,n] = (sum_k A[b,m,k]*Wt[n,k] + bias[n]) * scale
// One wave per block, one 16x16 tile. grid: x = C/16, y = ceil(M/16), z = b.
// ---------------------------------------------------------------------------
__global__ void __launch_bounds__(32) gemm_awt_kernel(
    const float* __restrict__ A, const float* __restrict__ Wt,
    const float* __restrict__ bias, float* __restrict__ Out,
    float scale, int M) {
  const int b = blockIdx.z;
  const float* Ab = A + (size_t)b * M * CC;
  float*       Ob = Out + (size_t)b * M * CC;

  const int n0   = blockIdx.x * 16;
  const int m0   = blockIdx.y * 16;
  const int lane = threadIdx.x & 31;
  const int half = lane >> 4;
  const int mr   = lane & 15;

  int arow = m0 + mr;
  if (arow >= M) arow = M - 1;  // clamp loads for partial tile; store is guarded

  v8f acc = {0.f,0.f,0.f,0.f,0.f,0.f,0.f,0.f};
  for (int k0 = 0; k0 < CC; k0 += 4) {
    v2f af = *(const v2f*)(Ab + (size_t)arow * CC + k0 + 2 * half);
    v2f bf = *(const v2f*)(Wt + (size_t)(n0 + mr) * CC + k0 + 2 * half);
    acc = wmma4(af, bf, acc);
  }

  const int n  = n0 + mr;
  const int mb = m0 + half * 8;
  const float bn = bias[n];
#pragma unroll
  for (int r = 0; r < 8; ++r) {
    const int row = mb + r;
    if (row < M) Ob[(size_t)row * CC + n] = (acc[r] + bn) * scale;
  }
}

// ---------------------------------------------------------------------------
// Kernel C: criss-cross sampling + softmax + weighted sum (per b,q,head).
// Block = 256 threads = 8 waves; wave w handles head w; lane = spatial pos.
// ---------------------------------------------------------------------------
__global__ void __launch_bounds__(256) attn_kernel(
    const float* __restrict__ pq, const float* __restrict__ pk,
    const float* __restrict__ pv, const float* __restrict__ pts,
    float* __restrict__ outp) {
  const int bq   = blockIdx.x;          // b*NQ + q
  const int b    = bq / NQ;
  const int head = threadIdx.x >> 5;
  const int lane = threadIdx.x & 31;

  const float x = pts[(size_t)bq * 2 + 0];
  const float y = pts[(size_t)bq * 2 + 1];
  float y0f = fminf(fmaxf(floorf(y), 0.f), (float)(HH - 1));
  const float wy = y - y0f;
  const int y0 = (int)y0f;
  const int y1 = min(y0 + 1, HH - 1);
  float x0f = fminf(fmaxf(floorf(x), 0.f), (float)(WW - 1));
  const float wx = x - x0f;
  const int x0 = (int)x0f;
  const int x1 = min(x0 + 1, WW - 1);
  const float oy = 1.f - wy, ox = 1.f - wx;

  const float* pqh = pq + (size_t)bq * CC + head * HD;
  const size_t cb  = ((size_t)b * CC + head * HD) * HWSZ;
  const float* pkh = pk + cb;
  const float* pvh = pv + cb;

  const float pqv = pqh[lane];  // lane = channel d for the pq broadcast source

  // Attention logits: lane covers positions {lane, lane+32} of rows (W) & cols (H)
  float wr0 = 0.f, wr1 = 0.f, wc0 = 0.f, wc1 = 0.f;
  for (int d = 0; d < HD; ++d) {
    const float s = __shfl(pqv, d, 32);
    const float* pr = pkh + (size_t)d * HWSZ;
    const float r0 = pr[y0 * WW + lane]        * oy + pr[y1 * WW + lane]        * wy;
    const float r1 = pr[y0 * WW + lane + 32]   * oy + pr[y1 * WW + lane + 32]   * wy;
    const float c0 = pr[lane * WW + x0]        * ox + pr[lane * WW + x1]        * wx;
    const float c1 = pr[(lane + 32) * WW + x0] * ox + pr[(lane + 32) * WW + x1] * wx;
    wr0 += s * r0; wr1 += s * r1; wc0 += s * c0; wc1 += s * c1;
  }

  // Softmax over the 128 concatenated logits (4 per lane x 32 lanes)
  float mx = fmaxf(fmaxf(wr0, wr1), fmaxf(wc0, wc1));
#pragma unroll
  for (int off = 16; off >= 1; off >>= 1)
    mx = fmaxf(mx, __shfl_xor(mx, off, 32));
  const float e0 = expf(wr0 - mx), e1 = expf(wr1 - mx);
  const float e2 = expf(wc0 - mx), e3 = expf(wc1 - mx);
  float sum = e0 + e1 + e2 + e3;
#pragma unroll
  for (int off = 16; off >= 1; off >>= 1)
    sum += __shfl_xor(sum, off, 32);
  const float inv = 1.f / sum;
  const float ar0 = e0 * inv, ar1 = e1 * inv, ac0 = e2 * inv, ac1 = e3 * inv;

  // out[d] = sum_w a_row[w]*v_row[d,w] + sum_k a_col[k]*v_col[d,k]
  float od = 0.f;
  for (int d = 0; d < HD; ++d) {
    const float* vr = pvh + (size_t)d * HWSZ;
    const float v0 = vr[y0 * WW + lane]        * oy + vr[y1 * WW + lane]        * wy;
    const float v1 = vr[y0 * WW + lane + 32]   * oy + vr[y1 * WW + lane + 32]   * wy;
    const float u0 = vr[lane * WW + x0]        * ox + vr[lane * WW + x1]        * wx;
    const float u1 = vr[(lane + 32) * WW + x0] * ox + vr[(lane + 32) * WW + x1] * wx;
    float p = ar0 * v0 + ar1 * v1 + ac0 * u0 + ac1 * u1;
#pragma unroll
    for (int off = 16; off >= 1; off >>= 1)
      p += __shfl_xor(p, off, 32);
    if (lane == d) od = p;   // after butterfly all lanes hold the total
  }
  outp[(size_t)bq * CC + head * HD + lane] = od;
}

// ---------------------------------------------------------------------------
extern "C" void kernel_launch(void* const* d_in, const int* in_sizes, int n_in,
                              void* d_out, int out_size, void* d_ws, size_t ws_size,
                              hipStream_t stream) {
  (void)in_sizes; (void)n_in; (void)out_size; (void)ws_size;
  const float* query = (const float*)d_in[0];
  const float* key   = (const float*)d_in[1];
  const float* value = (const float*)d_in[2];
  const float* pts   = (const float*)d_in[3];
  const float* Wq = (const float*)d_in[4];
  const float* bq = (const float*)d_in[5];
  const float* Wk = (const float*)d_in[6];
  const float* bk = (const float*)d_in[7];
  const float* Wv = (const float*)d_in[8];
  const float* bv = (const float*)d_in[9];
  const float* Wo = (const float*)d_in[10];
  const float* bo = (const float*)d_in[11];
  float* out = (float*)d_out;

  // Workspace layout: pk | pv | pq | pre  (~69 MB)
  float* pk  = (float*)d_ws;
  float* pv  = pk + (size_t)BB * CC * HWSZ;
  float* pq  = pv + (size_t)BB * CC * HWSZ;
  float* pre = pq + (size_t)BB * NQ * CC;

  const float SCALING = 0.17677669529663687f;  // 32^-0.5

  // K / V projections (WMMA f32)
  proj_khw_kernel<<<dim3(HWSZ / 64, CC / 16, BB * 2), 32, 0, stream>>>(
      Wk, bk, Wv, bv, key, value, pk, pv);

  // Q projection with scaling (WMMA f32)
  gemm_awt_kernel<<<dim3(CC / 16, (NQ + 15) / 16, BB), 32, 0, stream>>>(
      query, Wq, bq, pq, SCALING, NQ);

  // Criss-cross sampling + softmax + weighted sum
  attn_kernel<<<dim3(BB * NQ), 256, 0, stream>>>(pq, pk, pv, pts, pre);

  // Output projection (WMMA f32)
  gemm_awt_kernel<<<dim3(CC / 16, (NQ + 15) / 16, BB), 32, 0, stream>>>(
      pre, Wo, bo, out, 1.0f, NQ);
}